// PatchFeat_21990232555570
// MI455X (gfx1250) — compile-verified
//
#include <hip/hip_runtime.h>
#include <cstddef>

typedef __attribute__((ext_vector_type(2))) float v2f;
typedef __attribute__((ext_vector_type(8))) float v8f;

#define BN_EPS 1e-5f

// ============================================================================
// Transpose x (B,3,N) -> xt (B,N,3) and zero-padded GEMM input g0 (B*N,4)
// ============================================================================
__global__ void k_build_xt(const float* __restrict__ x, float* __restrict__ xt,
                           float* __restrict__ g0, int B, int N) {
  int total = B * N;
  for (int i = blockIdx.x * blockDim.x + threadIdx.x; i < total;
       i += gridDim.x * blockDim.x) {
    int b = i / N, n = i - b * N;
    float vx = x[((size_t)b * 3 + 0) * N + n];
    float vy = x[((size_t)b * 3 + 1) * N + n];
    float vz = x[((size_t)b * 3 + 2) * N + n];
    xt[(size_t)i * 3 + 0] = vx;
    xt[(size_t)i * 3 + 1] = vy;
    xt[(size_t)i * 3 + 2] = vz;
    g0[(size_t)i * 4 + 0] = vx;
    g0[(size_t)i * 4 + 1] = vy;
    g0[(size_t)i * 4 + 2] = vz;
    g0[(size_t)i * 4 + 3] = 0.f;
  }
}

// ============================================================================
// Weight pre-pack: W (kAct x cout) -> Wp, zero-padded to kStride rows and
// pair-interleaved as Wp[p][n][2] with p = k/2, so each lane's two K-values
// (k+kh, k+kh+1) are one aligned 8-byte load. Removes all guards from the
// WMMA inner loop.
// ============================================================================
__global__ void k_pack_w(const float* __restrict__ W, float* __restrict__ Wp,
                         int kAct, int kStride, int cout) {
  int pairs = kStride >> 1;
  int tot = pairs * cout;
  for (int i = blockIdx.x * blockDim.x + threadIdx.x; i < tot;
       i += gridDim.x * blockDim.x) {
    int p = i / cout, n = i - p * cout;
    int k0 = 2 * p, k1 = k0 + 1;
    Wp[(size_t)i * 2 + 0] = (k0 < kAct) ? W[(size_t)k0 * cout + n] : 0.f;
    Wp[(size_t)i * 2 + 1] = (k1 < kAct) ? W[(size_t)k1 * cout + n] : 0.f;
  }
}

// ============================================================================
// WMMA GEMM:  Z[rows x cout] = G[rows x kStride] * W + bias, f32 end-to-end
// via V_WMMA_F32_16X16X4_F32. One 16x32 tile per wave (A fragment reused by
// two independent accumulators), 8 waves per block.
// A layout (ISA 7.12.2): lanes 0-15 -> K{0,1}, lanes 16-31 -> K{2,3}, M=lane&15.
// C/D layout: VGPR v -> M = v + 8*(lane>=16), N = lane&15.
// G rows and packed W are zero-padded to kStride, so no guards needed.
// ============================================================================
__global__ void k_gemm_wmma(const float* __restrict__ G, const float* __restrict__ Wp,
                            const float* __restrict__ bias, float* __restrict__ Z,
                            int rows, int kStride, int cout, int relu) {
  int tilesM = rows >> 4;
  int tilesN = cout >> 5;                          // 32-wide N tiles
  int wave = (blockIdx.x * blockDim.x + threadIdx.x) >> 5;
  if (wave >= tilesM * tilesN) return;             // wave-uniform: EXEC all-1s
  int mT = wave % tilesM;
  int nT = wave / tilesM;
  int lane = threadIdx.x & 31;
  int l16  = lane & 15;
  int hi   = lane >> 4;
  int m  = mT * 16 + l16;
  int n0 = nT * 32 + l16;                          // second tile at n0+16

  const v2f* ga = (const v2f*)(G + (size_t)m * kStride) + hi;     // (k/2 + hi)
  const v2f* wp = (const v2f*)Wp + (size_t)hi * cout + n0;        // pair p, col n0

  v8f acc0 = {0.f, 0.f, 0.f, 0.f, 0.f, 0.f, 0.f, 0.f};
  v8f acc1 = {0.f, 0.f, 0.f, 0.f, 0.f, 0.f, 0.f, 0.f};
#pragma unroll 2
  for (int k = 0; k < kStride; k += 4) {
    v2f a  = *ga;                                  // global_load_b64
    v2f b0 = wp[0];                                // global_load_b64
    v2f b1 = wp[16];                               // global_load_b64
    __builtin_prefetch(ga + 8, 0, 3);              // global_prefetch_b8
    acc0 = __builtin_amdgcn_wmma_f32_16x16x4_f32(
        false, a, false, b0, (short)0, acc0, false, false);
    acc1 = __builtin_amdgcn_wmma_f32_16x16x4_f32(
        false, a, false, b1, (short)0, acc1, false, false);
    ga += 2;                                       // k advances 4 -> 2 pairs
    wp += 2 * (size_t)cout;                        // p advances 2 rows of pairs
  }

  float bv0 = bias[n0];
  float bv1 = bias[n0 + 16];
  int mBase = mT * 16 + hi * 8;
#pragma unroll
  for (int v = 0; v < 8; ++v) {
    float* zr = Z + (size_t)(mBase + v) * cout;
    float z0 = acc0[v] + bv0;
    float z1 = acc1[v] + bv1;
    if (relu) { z0 = fmaxf(z0, 0.f); z1 = fmaxf(z1, 0.f); }
    zr[n0]      = z0;
    zr[n0 + 16] = z1;
  }
}

// ============================================================================
// BN statistics: per-channel sum / sumsq via LDS atomics -> global atomics
// ============================================================================
__global__ void k_zero(float* __restrict__ p, int n) {
  for (int i = blockIdx.x * blockDim.x + threadIdx.x; i < n;
       i += gridDim.x * blockDim.x)
    p[i] = 0.f;
}

__global__ void k_stats(const float* __restrict__ X, float* __restrict__ st,
                        int rows, int C) {
  __shared__ float s0[1024];
  __shared__ float s1[1024];
  for (int c = threadIdx.x; c < C; c += blockDim.x) { s0[c] = 0.f; s1[c] = 0.f; }
  __syncthreads();
  size_t total = (size_t)rows * C;
  for (size_t e = blockIdx.x * (size_t)blockDim.x + threadIdx.x; e < total;
       e += (size_t)gridDim.x * blockDim.x) {
    int c = (int)(e % C);
    float v = X[e];
    atomicAdd(&s0[c], v);
    atomicAdd(&s1[c], v * v);
  }
  __syncthreads();
  for (int c = threadIdx.x; c < C; c += blockDim.x) {
    atomicAdd(&st[c], s0[c]);
    atomicAdd(&st[C + c], s1[c]);
  }
}

// y = gamma*(x-mean)*rsqrt(var+eps)+beta  (+ optional trailing ReLU)
__global__ void k_bn(const float* __restrict__ X, float* __restrict__ Y,
                     const float* __restrict__ st, const float* __restrict__ gm,
                     const float* __restrict__ bt, int rows, int C, int relu) {
  size_t total = (size_t)rows * C;
  float inv = 1.0f / (float)rows;
  for (size_t e = blockIdx.x * (size_t)blockDim.x + threadIdx.x; e < total;
       e += (size_t)gridDim.x * blockDim.x) {
    int c = (int)(e % C);
    float mean = st[c] * inv;
    float var  = st[C + c] * inv - mean * mean;
    float y = gm[c] * (X[e] - mean) * rsqrtf(var + BN_EPS) + bt[c];
    if (relu) y = fmaxf(y, 0.f);
    Y[e] = y;
  }
}

// ============================================================================
// Farthest point sampling: one workgroup per batch; dist[] in LDS; argmax
// tree-reduce with lowest-index tie break (matches jnp.argmax).
// First emitted index is 0 (matches lax.scan semantics in the reference).
// ============================================================================
__global__ void k_fps(const float* __restrict__ xyz, int N, int npoint,
                      int* __restrict__ out) {
  __shared__ float dist[4096];
  __shared__ float rv[256];
  __shared__ int   ri[256];
  __shared__ int   sFar;
  int b = blockIdx.x, t = threadIdx.x;
  const float* xb = xyz + (size_t)b * N * 3;
  for (int i = t; i < N; i += blockDim.x) dist[i] = 1e10f;
  if (t == 0) sFar = 0;
  __syncthreads();
  for (int it = 0; it < npoint; ++it) {
    int f = sFar;
    if (t == 0) out[b * npoint + it] = f;
    float cx = xb[f * 3 + 0], cy = xb[f * 3 + 1], cz = xb[f * 3 + 2];
    float bv = -1.f; int bi = 0x7fffffff;
    for (int i = t; i < N; i += blockDim.x) {
      float dx = xb[i * 3 + 0] - cx;
      float dy = xb[i * 3 + 1] - cy;
      float dz = xb[i * 3 + 2] - cz;
      float d  = dx * dx + dy * dy + dz * dz;
      float nd = fminf(dist[i], d);
      dist[i] = nd;
      if (nd > bv) { bv = nd; bi = i; }
    }
    rv[t] = bv; ri[t] = bi;
    __syncthreads();
    for (int s = blockDim.x >> 1; s > 0; s >>= 1) {
      if (t < s) {
        float ov = rv[t + s]; int oi = ri[t + s];
        if (ov > rv[t] || (ov == rv[t] && oi < ri[t])) { rv[t] = ov; ri[t] = oi; }
      }
      __syncthreads();
    }
    if (t == 0) sFar = ri[0];
    __syncthreads();
  }
}

__global__ void k_gather_xyz(const float* __restrict__ xyz, const int* __restrict__ idx,
                             float* __restrict__ outp, int Nin, int M, int B) {
  int tot = B * M;
  for (int i = blockIdx.x * blockDim.x + threadIdx.x; i < tot;
       i += gridDim.x * blockDim.x) {
    int b = i / M;
    int s = idx[i];
    const float* p = xyz + ((size_t)b * Nin + s) * 3;
    outp[(size_t)i * 3 + 0] = p[0];
    outp[(size_t)i * 3 + 1] = p[1];
    outp[(size_t)i * 3 + 2] = p[2];
  }
}

// ============================================================================
// Query-ball + gather + concat([feats, xyz-center]) with zero pad to cinPad.
// First-K in ascending index order (== reference's sort of masked arange),
// short groups padded with the first in-ball index.
// ============================================================================
__global__ void k_group(const float* __restrict__ xyz, const float* __restrict__ nxyz,
                        const float* __restrict__ feats, int Nin, int M, int Cf,
                        int K, float r2, float* __restrict__ out, int cinPad) {
  __shared__ int scnt[128];
  __shared__ int soff[128];
  __shared__ int sel[128];
  __shared__ int sTot;
  int g = blockIdx.x;
  int b = g / M;
  int t = threadIdx.x;
  float cx = nxyz[(size_t)g * 3 + 0];
  float cy = nxyz[(size_t)g * 3 + 1];
  float cz = nxyz[(size_t)g * 3 + 2];
  int chunk = (Nin + 127) / 128;
  int i0 = t * chunk;
  int i1 = i0 + chunk; if (i1 > Nin) i1 = Nin;
  const float* xb = xyz + (size_t)b * Nin * 3;

  int c = 0;
  for (int i = i0; i < i1; ++i) {
    float dx = xb[i * 3 + 0] - cx, dy = xb[i * 3 + 1] - cy, dz = xb[i * 3 + 2] - cz;
    if (dx * dx + dy * dy + dz * dz <= r2) ++c;
  }
  scnt[t] = c;
  __syncthreads();
  if (t == 0) {
    int acc = 0;
    for (int j = 0; j < 128; ++j) { soff[j] = acc; acc += scnt[j]; }
    sTot = acc;
  }
  __syncthreads();
  int pos = soff[t];
  for (int i = i0; i < i1 && pos < K; ++i) {
    float dx = xb[i * 3 + 0] - cx, dy = xb[i * 3 + 1] - cy, dz = xb[i * 3 + 2] - cz;
    if (dx * dx + dy * dy + dz * dz <= r2) { sel[pos] = i; ++pos; }
  }
  __syncthreads();
  int T = sTot;                       // >= 1: centroid is its own neighbor
  int first = sel[0];
  for (int p = t; p < K; p += 128)
    if (p >= T) sel[p] = first;
  __syncthreads();

  int total = K * cinPad;
  float* orow = out + (size_t)g * K * cinPad;
  for (int e = t; e < total; e += 128) {
    int kk = e / cinPad, cc = e - kk * cinPad;
    int si = sel[kk];
    float v;
    if (cc < Cf) {
      v = feats[((size_t)b * Nin + si) * Cf + cc];
    } else if (cc < Cf + 3) {
      int a = cc - Cf;
      float ctr = (a == 0) ? cx : (a == 1) ? cy : cz;
      v = xb[si * 3 + a] - ctr;
    } else {
      v = 0.f;                         // zero pad -> safe WMMA K loop
    }
    orow[e] = v;
  }
}

// max over K rows per group; write into concat slot [coff, coff+C)
__global__ void k_maxpool(const float* __restrict__ X, int K, int C,
                          float* __restrict__ out, int Ctot, int coff) {
  int g = blockIdx.x;
  const float* xg = X + (size_t)g * K * C;
  float* og = out + (size_t)g * Ctot + coff;
  for (int c = threadIdx.x; c < C; c += blockDim.x) {
    float mv = xg[c];
    for (int kk = 1; kk < K; ++kk) mv = fmaxf(mv, xg[(size_t)kk * C + c]);
    og[c] = mv;
  }
}

// SA3 group_all input: rows = B*64, cols 644 = [l2_xyz(3) | l2_pts(640) | pad]
__global__ void k_sa3build(const float* __restrict__ nx2, const float* __restrict__ l2p,
                           float* __restrict__ out) {
  const int ROWS = 8 * 64, CP = 644;
  int tot = ROWS * CP;
  for (int e = blockIdx.x * blockDim.x + threadIdx.x; e < tot;
       e += gridDim.x * blockDim.x) {
    int r = e / CP, c = e - r * CP;
    float v;
    if (c < 3)        v = nx2[(size_t)r * 3 + c];
    else if (c < 643) v = l2p[(size_t)r * 640 + (c - 3)];
    else              v = 0.f;
    out[e] = v;
  }
}

// out[b][c][n] = l3[b][c]  (float4 stores)
__global__ void k_broadcast(const float* __restrict__ l3, float4* __restrict__ out,
                            size_t nq) {
  for (size_t i = blockIdx.x * (size_t)blockDim.x + threadIdx.x; i < nq;
       i += (size_t)gridDim.x * blockDim.x) {
    float v = l3[(i * 4) >> 12];       // 4096 floats per (b,c) row
    float4 r; r.x = v; r.y = v; r.z = v; r.w = v;
    out[i] = r;
  }
}

// ============================================================================
// Host orchestration
// ============================================================================
extern "C" void kernel_launch(void* const* d_in, const int* in_sizes, int n_in,
                              void* d_out, int out_size, void* d_ws, size_t ws_size,
                              hipStream_t stream) {
  (void)in_sizes; (void)n_in; (void)out_size; (void)ws_size;
  const int B = 8, N = 4096;

  // ---- inputs, setup_inputs() insertion order; tuples are (W, b, gamma, beta)
  int ii = 0;
  auto F = [&](void) { return (const float*)d_in[ii++]; };
  const float* x    = F();
  const float* c1w  = F(); const float* c1b  = F();
  const float* bn1g = F(); const float* bn1b = F();
  const float* c2w  = F(); const float* c2b  = F();
  const float* bn2g = F(); const float* bn2b = F();
  const float* P1[3][3][4];
  for (int s = 0; s < 3; ++s) for (int l = 0; l < 3; ++l) for (int t = 0; t < 4; ++t) P1[s][l][t] = F();
  const float* P2[3][3][4];
  for (int s = 0; s < 3; ++s) for (int l = 0; l < 3; ++l) for (int t = 0; t < 4; ++t) P2[s][l][t] = F();
  const float* P3[3][4];
  for (int l = 0; l < 3; ++l) for (int t = 0; t < 4; ++t) P3[l][t] = F();

  // ---- workspace carve-up (floats, 256B aligned)
  float* ws = (float*)d_ws;
  size_t o = 0;
  auto alloc = [&](size_t nf) { float* p = ws + o; o += (nf + 63) & ~(size_t)63; return p; };
  float* xt    = alloc((size_t)B * N * 3);
  float* g0    = alloc((size_t)B * N * 4);
  float* h     = alloc((size_t)B * N * 128);
  float* stats = alloc(2048);
  float* wpad  = alloc((size_t)512 * 1024);    // packed weights (max 512x1024 >= all)
  int*   fps1  = (int*)alloc(B * 128);
  float* nx1   = alloc((size_t)B * 128 * 3);
  float* l1p   = alloc((size_t)B * 128 * 320);
  int*   fps2  = (int*)alloc(B * 64);
  float* nx2   = alloc((size_t)B * 64 * 3);
  float* l2p   = alloc((size_t)B * 64 * 640);
  float* l3    = alloc((size_t)B * 1024);
  float* bufA  = alloc((size_t)65536 * 324);   // 85 MB: gathered groups / ping
  float* bufB  = alloc((size_t)131072 * 128);  // 67 MB: pong

  // ---- one MLP layer: pack W -> GEMM (+opt pre-BN ReLU) -> stats -> BN (+ReLU)
  auto run_layer = [&](const float* in, const float* W, const float* bb,
                       const float* gm, const float* be, int rows, int kAct,
                       int kStride, int cout, bool convStyle, float* zbuf,
                       float* ybuf) {
    int wtot = (kStride >> 1) * cout;
    k_pack_w<<<(wtot + 255) / 256 < 256 ? (wtot + 255) / 256 : 256, 256, 0,
               stream>>>(W, wpad, kAct, kStride, cout);
    int tiles = (rows / 16) * (cout / 32);
    k_gemm_wmma<<<(tiles + 7) / 8, 256, 0, stream>>>(in, wpad, bb, zbuf, rows,
                                                     kStride, cout,
                                                     convStyle ? 1 : 0);
    k_zero<<<8, 256, 0, stream>>>(stats, 2 * cout);
    k_stats<<<256, 256, 0, stream>>>(zbuf, stats, rows, cout);
    size_t tot = (size_t)rows * cout;
    int blk = (int)((tot + 255) / 256); if (blk > 4096) blk = 4096;
    k_bn<<<blk, 256, 0, stream>>>(zbuf, ybuf, stats, gm, be, rows, cout,
                                  convStyle ? 0 : 1);
  };

  // ---- stem: conv1 -> conv2 (ReLU before BN, per reference)
  k_build_xt<<<256, 256, 0, stream>>>(x, xt, g0, B, N);
  run_layer(g0,   c1w, c1b, bn1g, bn1b, B * N,  3, 4,  64, true, bufB, bufB);
  run_layer(bufB, c2w, c2b, bn2g, bn2b, B * N, 64, 64, 128, true, bufA, h);

  // ---- multi-scale set-abstraction stage
  auto run_sa = [&](const float* xyz, int Nin, const float* feats, int Cf, int M,
                    const float* r2, const int* Ks, const float* (*P)[3][4],
                    const int (*CH)[3], int cinPad, int* fpsIdx, float* nx,
                    float* lout, int Ctot) {
    k_fps<<<B, 256, 0, stream>>>(xyz, Nin, M, fpsIdx);
    k_gather_xyz<<<(B * M + 255) / 256, 256, 0, stream>>>(xyz, fpsIdx, nx, Nin, M, B);
    int coff = 0;
    for (int s = 0; s < 3; ++s) {
      int K = Ks[s], rows = B * M * K;
      int c0 = CH[s][0], c1 = CH[s][1], c2 = CH[s][2];
      k_group<<<B * M, 128, 0, stream>>>(xyz, nx, feats, Nin, M, Cf, K, r2[s],
                                         bufA, cinPad);
      run_layer(bufA, P[s][0][0], P[s][0][1], P[s][0][2], P[s][0][3],
                rows, Cf + 3, cinPad, c0, false, bufB, bufB);
      run_layer(bufB, P[s][1][0], P[s][1][1], P[s][1][2], P[s][1][3],
                rows, c0, c0, c1, false, bufA, bufA);
      run_layer(bufA, P[s][2][0], P[s][2][1], P[s][2][2], P[s][2][3],
                rows, c1, c1, c2, false, bufB, bufB);
      k_maxpool<<<B * M, 128, 0, stream>>>(bufB, K, c2, lout, Ctot, coff);
      coff += c2;
    }
  };

  const float r2a[3] = {0.01f * 0.01f, 0.02f * 0.02f, 0.04f * 0.04f};
  const float r2b[3] = {0.02f * 0.02f, 0.04f * 0.04f, 0.08f * 0.08f};
  const int   Ka[3]  = {16, 32, 128};
  const int   Kb[3]  = {32, 64, 128};
  const int   CHa[3][3] = {{64, 64, 64}, {64, 64, 128}, {64, 96, 128}};
  const int   CHb[3][3] = {{64, 64, 128}, {128, 128, 256}, {128, 128, 256}};

  run_sa(xt,  4096, h,   128, 128, r2a, Ka, P1, CHa, 132, fps1, nx1, l1p, 320);
  run_sa(nx1, 128,  l1p, 320, 64,  r2b, Kb, P2, CHb, 324, fps2, nx2, l2p, 640);

  // ---- SA3 group_all: concat([l2_xyz, l2_pts]) -> MLP(256,512,1024) -> max
  k_sa3build<<<512, 256, 0, stream>>>(nx2, l2p, bufA);
  run_layer(bufA, P3[0][0], P3[0][1], P3[0][2], P3[0][3], 512, 643, 644,  256, false, bufB, bufB);
  run_layer(bufB, P3[1][0], P3[1][1], P3[1][2], P3[1][3], 512, 256, 256,  512, false, bufA, bufA);
  run_layer(bufA, P3[2][0], P3[2][1], P3[2][2], P3[2][3], 512, 512, 512, 1024, false, bufB, bufB);
  k_maxpool<<<B, 256, 0, stream>>>(bufB, 64, 1024, l3, 1024, 0);

  // ---- broadcast (B,1024) -> (B,1024,4096)
  size_t nq = (size_t)B * 1024 * 4096 / 4;
  k_broadcast<<<4096, 256, 0, stream>>>(l3, (float4*)d_out, nq);
}